// Quantizer_18141941858842
// MI455X (gfx1250) — compile-verified
//
#include <hip/hip_runtime.h>
#include <math.h>

typedef __attribute__((ext_vector_type(2))) float v2f;
typedef __attribute__((ext_vector_type(8))) float v8f;
typedef __attribute__((ext_vector_type(4))) unsigned int u32x4;
typedef __attribute__((ext_vector_type(8))) int i32x8;
typedef __attribute__((ext_vector_type(4))) int i32x4;

#define NTOT   131072            // 32*64*64 flattened rows
#define DDIM   64
#define KCB    512
#define CHSTR  4096              // element stride between channels (NCHW)
#define IMSTR  262144            // 64*4096 elements per batch image

// ws layout (floats): [0] = sse accumulator, [1..512] = e_norm, [513..1024] = counts

__global__ __launch_bounds__(512) void vq_init(const float* __restrict__ e,
                                               float* __restrict__ ws) {
    int k = threadIdx.x;                // 0..511
    const float* row = e + k * DDIM;
    float s = 0.f;
#pragma unroll
    for (int d = 0; d < DDIM; ++d) { float v = row[d]; s += v * v; }
    ws[1 + k]   = s;                    // e_norm
    ws[513 + k] = 0.f;                  // counts
    if (k == 0) ws[0] = 0.f;            // sse
}

// TDM: load one 16x64-f32 codebook tile (4 KB, contiguous) into LDS.
// D# built per CDNA5 ISA ch.8: group0 = {count=1 | lds_addr | global_addr | type=2},
// group1 = {mask=0, data_size=4B, tensor 64x512, tile 64x16, stride 64}.
__device__ __forceinline__ void tdm_load_tile(const float* gsrc, unsigned lds_off) {
    const unsigned long long ga = (unsigned long long)(uintptr_t)gsrc;
    u32x4 g0;
    g0.x = 1u;                                           // count=1, user D#
    g0.y = lds_off;                                      // LDS byte address
    g0.z = (unsigned)(ga & 0xffffffffu);                 // global_addr[31:0]
    g0.w = (unsigned)((ga >> 32) & 0x1ffffffu) | (2u << 30);  // addr[56:32] | type=2
    i32x8 g1;
    g1[0] = (int)(2u << 16);      // workgroup_mask=0, data_size=2 (4 bytes)
    g1[1] = (int)(64u << 16);     // tensor_dim0=64  (bits 79:48, low half)
    g1[2] = (int)(512u << 16);    // tensor_dim1=512 (bits 111:80, low half)
    g1[3] = (int)(64u << 16);     // tile_dim0=64    (bits 127:112)
    g1[4] = 16;                   // tile_dim1=16, tile_dim2=0
    g1[5] = 64;                   // tensor_dim0_stride=64 (rows contiguous)
    g1[6] = 0;
    g1[7] = 0;
    i32x4 gz; gz[0] = 0; gz[1] = 0; gz[2] = 0; gz[3] = 0;   // groups 2/3 unused (2D)
#if __clang_major__ >= 23
    i32x8 gz8; gz8[0]=0; gz8[1]=0; gz8[2]=0; gz8[3]=0; gz8[4]=0; gz8[5]=0; gz8[6]=0; gz8[7]=0;
    __builtin_amdgcn_tensor_load_to_lds(g0, g1, gz, gz, gz8, 0);
#else
    __builtin_amdgcn_tensor_load_to_lds(g0, g1, gz, gz, 0);
#endif
}

__global__ __launch_bounds__(256) void vq_main(const float* __restrict__ z,
                                               const float* __restrict__ e,
                                               float* __restrict__ out_zq,   // d_out + 1
                                               float* __restrict__ ws) {
    __shared__ float sB[2][16 * DDIM];             // double-buffered B tiles (8 KB)
    __shared__ int sIdx[8][32];                    // winning code per row, per wave

    const int lane = threadIdx.x & 31;
    const int wave = threadIdx.x >> 5;
    const int r0 = (blockIdx.x * 8 + wave) * 32;   // 32 rows per wave (2 fragments)
    const int half = lane >> 4;                    // C-layout half: rows 0-7 vs 8-15
    const int l15  = lane & 15;
    const int off  = half * 2;                     // K sub-offset per ISA A/B layout

    // ---- Load two A fragments (rows r0..r0+31); accumulate ||z||^2 partials ----
    v2f a[2][16];
    float zn[2];                                   // ||z_row||^2 for row r0+f*16+l15
#pragma unroll
    for (int f = 0; f < 2; ++f) {
        const int rowA = r0 + f * 16 + l15;
        const int bA   = rowA >> 12;               // batch
        const int hwA  = rowA & 4095;              // h*64+w
        const float* zrow = z + (size_t)bA * IMSTR + hwA;
        float p = 0.f;
#pragma unroll
        for (int kk = 0; kk < 16; ++kk) {
            const int kb = kk * 4 + off;
            const float x = zrow[(size_t)kb * CHSTR];
            const float y = zrow[(size_t)(kb + 1) * CHSTR];
            a[f][kk].x = x;
            a[f][kk].y = y;
            p += x * x + y * y;
        }
        zn[f] = p + __shfl_xor(p, 16, 32);         // other 32 K-elems live in lane^16
    }

    const float* enorm = ws + 1;
    float bestS[2][8];
    int   bestI[2][8];
#pragma unroll
    for (int f = 0; f < 2; ++f)
#pragma unroll
        for (int j = 0; j < 8; ++j) { bestS[f][j] = 3.4e38f; bestI[f][j] = 0; }

    const unsigned ldsB0 = (unsigned)(uintptr_t)&sB[0][0];   // LDS byte offset
    const unsigned ldsB1 = (unsigned)(uintptr_t)&sB[1][0];

    // ---- Prologue: TDM tile 0 into buffer 0 ----
    if (wave == 0) {
        tdm_load_tile(e, ldsB0);
        __builtin_amdgcn_s_wait_tensorcnt(0);
    }
    __syncthreads();

    // ---- Sweep 512 codes in 16-column tiles; TDM prefetch overlapped with WMMA ----
    for (int ct = 0; ct < 32; ++ct) {
        const int cur = ct & 1;
        if (wave == 0 && ct < 31)                  // prefetch next tile into other buffer
            tdm_load_tile(e + (ct + 1) * (16 * DDIM), cur ? ldsB0 : ldsB1);

        const float* btile = &sB[cur][0];
        v8f acc0 = {}, acc1 = {};
#pragma unroll
        for (int kk = 0; kk < 16; ++kk) {
            const int kb = kk * 4 + off;
            const v2f bb = *(const v2f*)&btile[l15 * DDIM + kb];   // ds_load_b64
            acc0 = __builtin_amdgcn_wmma_f32_16x16x4_f32(
                false, a[0][kk], false, bb, (short)0, acc0, false, false);
            acc1 = __builtin_amdgcn_wmma_f32_16x16x4_f32(
                false, a[1][kk], false, bb, (short)0, acc1, false, false);
        }
        const int n = ct * 16 + l15;               // code index for this lane's column
        const float en = enorm[n];
#pragma unroll
        for (int j = 0; j < 8; ++j) {
            const float s0 = en - 2.0f * acc0[j];  // ||e||^2 - 2 z.e
            if (s0 < bestS[0][j]) { bestS[0][j] = s0; bestI[0][j] = n; }
            const float s1 = en - 2.0f * acc1[j];
            if (s1 < bestS[1][j]) { bestS[1][j] = s1; bestI[1][j] = n; }
        }

        if (wave == 0 && ct < 31)
            __builtin_amdgcn_s_wait_tensorcnt(0);  // prefetch done before barrier
        __syncthreads();
    }

    // ---- Argmin reduction across 16 lanes within each half (xor masks < 16) ----
#pragma unroll
    for (int m = 1; m < 16; m <<= 1) {
#pragma unroll
        for (int f = 0; f < 2; ++f)
#pragma unroll
            for (int j = 0; j < 8; ++j) {
                const float os = __shfl_xor(bestS[f][j], m, 32);
                const int   oi = __shfl_xor(bestI[f][j], m, 32);
                if (os < bestS[f][j] || (os == bestS[f][j] && oi < bestI[f][j])) {
                    bestS[f][j] = os; bestI[f][j] = oi;
                }
            }
    }

    // ---- SSE from scores: ||z-e||^2 = ||z||^2 + bestScore (no z re-read) ----
    float sse = 0.f;
#pragma unroll
    for (int f = 0; f < 2; ++f) {
#pragma unroll
        for (int j = 0; j < 8; ++j) {
            // znorm of row r0+f*16+half*8+j lives in lane (half*8+j); all lanes active
            const float znr = __shfl(zn[f], half * 8 + j, 32);
            if (l15 == 0) {
                sse += znr + bestS[f][j];
                atomicAdd(ws + 513 + bestI[f][j], 1.0f);   // usage counts
                sIdx[wave][f * 16 + half * 8 + j] = bestI[f][j];
            }
        }
    }
#pragma unroll
    for (int m = 1; m < 32; m <<= 1) sse += __shfl_xor(sse, m, 32);
    if (lane == 0) atomicAdd(ws, sse);

    __syncthreads();

    // ---- Coalesced z_q scatter: lane <-> hw, loop over channels ----
    const int myIdx = sIdx[wave][lane];            // winning code for row r0+lane
    const float* crow = e + myIdx * DDIM;          // contiguous 256B gather (L2-hot)
    const int b  = r0 >> 12;
    const int hw = (r0 & 4095) + lane;             // 32 consecutive hw per wave
    float* obase = out_zq + (size_t)b * IMSTR + hw;
#pragma unroll
    for (int d = 0; d < DDIM; ++d) {
        obase[(size_t)d * CHSTR] = crow[d];        // 128B cacheline per store instr
    }
}

__global__ __launch_bounds__(512) void vq_final(const float* __restrict__ ws,
                                                float* __restrict__ d_out) {
    __shared__ float red[512];
    const int k = threadIdx.x;
    const float em = ws[513 + k] / (float)NTOT;
    const float t  = em + 1e-10f;
    red[k] = t * logf(t);
    __syncthreads();
    for (int s = 256; s > 0; s >>= 1) {
        if (k < s) red[k] += red[k + s];
        __syncthreads();
    }
    if (k == 0) {
        d_out[0] = 1.25f * ws[0] / (float)((size_t)NTOT * DDIM);   // (1+beta)*mse
        d_out[1 + (size_t)NTOT * DDIM] = expf(-red[0]);            // perplexity
    }
}

extern "C" void kernel_launch(void* const* d_in, const int* in_sizes, int n_in,
                              void* d_out, int out_size, void* d_ws, size_t ws_size,
                              hipStream_t stream) {
    (void)in_sizes; (void)n_in; (void)out_size; (void)ws_size;
    const float* z = (const float*)d_in[0];
    const float* e = (const float*)d_in[1];
    float* out = (float*)d_out;
    float* ws  = (float*)d_ws;

    vq_init<<<1, 512, 0, stream>>>(e, ws);
    vq_main<<<NTOT / 256, 256, 0, stream>>>(z, e, out + 1, ws);
    vq_final<<<1, 512, 0, stream>>>(ws, out);
}